// CombinedLoss_21990232555881
// MI455X (gfx1250) — compile-verified
//
#include <hip/hip_runtime.h>
#include <math.h>

// ---------------------------------------------------------------------------
// CombinedLoss (LCCNet-style) for MI455X / gfx1250.
//
// Memory-bound: point_clouds = 32*4*200000 f32 = 102.4 MB; reading only the
// x/y/z rows (w==1 by construction, and row-3 of the rigid inverse is exact)
// gives a 76.8 MB stream -> ~3.3 us HBM floor @ 23.3 TB/s (less on replay,
// since it fits the 192 MB L2; hence regular-temporal loads, not NT).
//
// All matrix algebra is folded analytically into one 4x4 "A" per batch:
//   RT_inv = RT_pred^-1 * RT_target = [Rp^T Rt | Rp^T (t_t - t_p)]
//   A      = RT_inv - diag(1,1,1,0), zero bottom row
// so the per-point error is ||A . p||.  The 4x4 * 4x16 tile product maps onto
// V_WMMA_F32_16X16X4_F32 (A rows 4..15 zero => upper-lane D fragments are
// exact zeros, so the sqrt-accumulate needs no masking).
//
// Loop control is forced scalar (readfirstlane on the wave id) so the hot
// loop has no EXEC-mask manipulation; the ragged tail (never taken for
// N=200000) is a separate scalar-branched epilogue with EXEC all-ones.
// ---------------------------------------------------------------------------

typedef float v2f __attribute__((ext_vector_type(2)));
typedef float v8f __attribute__((ext_vector_type(8)));

#ifndef __has_builtin
#define __has_builtin(x) 0
#endif
#if __has_builtin(__builtin_amdgcn_wmma_f32_16x16x4_f32)
#define USE_WMMA_F32X4 1
#else
#define USE_WMMA_F32X4 0
#endif

#define GRID_X 32  // blocks per batch for the streaming kernel

__device__ __forceinline__ float fast_sqrtf(float x) {
#if __has_builtin(__builtin_amdgcn_sqrtf)
    return __builtin_amdgcn_sqrtf(x);   // bare v_sqrt_f32 (~1 ulp)
#else
    return sqrtf(x);
#endif
}

__device__ __forceinline__ void quat_to_R(const float q[4], float R[3][3]) {
    float n2 = q[0]*q[0] + q[1]*q[1] + q[2]*q[2] + q[3]*q[3];
    float inv = rsqrtf(n2);
    float w = q[0]*inv, x = q[1]*inv, y = q[2]*inv, z = q[3]*inv;
    R[0][0] = 1.0f - 2.0f*y*y - 2.0f*z*z;
    R[0][1] = 2.0f*x*y - 2.0f*z*w;
    R[0][2] = 2.0f*x*z + 2.0f*y*w;
    R[1][0] = 2.0f*x*y + 2.0f*z*w;
    R[1][1] = 1.0f - 2.0f*x*x - 2.0f*z*z;
    R[1][2] = 2.0f*y*z - 2.0f*x*w;
    R[2][0] = 2.0f*x*z - 2.0f*y*w;
    R[2][1] = 2.0f*y*z + 2.0f*x*w;
    R[2][2] = 1.0f - 2.0f*x*x - 2.0f*y*y;
}

// ---------------------------------------------------------------------------
// Kernel 1: one wave.  Lane b handles batch b: smooth-L1 translation loss,
// quaternion angular distance, and the folded matrix A[b] (16 floats).
// ---------------------------------------------------------------------------
__global__ void k_pose_and_mats(const float* __restrict__ tgt_t,
                                const float* __restrict__ tgt_q,
                                const float* __restrict__ prd_t,
                                const float* __restrict__ prd_q,
                                float* __restrict__ wsA,     // [B][16]
                                float* __restrict__ wsS,     // [2]: loss_transl, loss_rot
                                int B) {
    const int b = threadIdx.x;
    float st = 0.0f, rdeg = 0.0f;
    if (b < B) {
        float tt[3], tp[3], qt[4], qp[4];
        #pragma unroll
        for (int i = 0; i < 3; ++i) { tt[i] = tgt_t[b*3 + i]; tp[i] = prd_t[b*3 + i]; }
        #pragma unroll
        for (int i = 0; i < 4; ++i) { qt[i] = tgt_q[b*4 + i]; qp[i] = prd_q[b*4 + i]; }

        // smooth_l1(transl_err, target_transl).sum()
        #pragma unroll
        for (int i = 0; i < 3; ++i) {
            float d = fabsf(tp[i] - tt[i]);
            st += (d < 1.0f) ? 0.5f * d * d : d - 0.5f;
        }

        // quaternion_distance(rot_err=qp, target_rot=qt)  (scale-invariant)
        float r0 = qt[0], r1 = -qt[1], r2 = -qt[2], r3 = -qt[3];
        float hw = qp[0]*r0 - qp[1]*r1 - qp[2]*r2 - qp[3]*r3;
        float hx = qp[0]*r1 + qp[1]*r0 + qp[2]*r3 - qp[3]*r2;
        float hy = qp[0]*r2 - qp[1]*r3 + qp[2]*r0 + qp[3]*r1;
        float hz = qp[0]*r3 + qp[1]*r2 - qp[2]*r1 + qp[3]*r0;
        float ang = 2.0f * atan2f(sqrtf(hx*hx + hy*hy + hz*hz), fabsf(hw));
        rdeg = ang * (180.0f / 3.14159265358979323846f);

        // A = RT_pred^-1 * RT_target - diag(1,1,1,0)
        float Rt[3][3], Rp[3][3];
        quat_to_R(qt, Rt);
        quat_to_R(qp, Rp);
        float dt[3] = { tt[0]-tp[0], tt[1]-tp[1], tt[2]-tp[2] };

        float Amat[16];
        #pragma unroll
        for (int i = 0; i < 3; ++i) {
            #pragma unroll
            for (int k = 0; k < 3; ++k) {
                float m = Rp[0][i]*Rt[0][k] + Rp[1][i]*Rt[1][k] + Rp[2][i]*Rt[2][k];
                Amat[i*4 + k] = m - ((i == k) ? 1.0f : 0.0f);
            }
            Amat[i*4 + 3] = Rp[0][i]*dt[0] + Rp[1][i]*dt[1] + Rp[2][i]*dt[2];
        }
        Amat[12] = Amat[13] = Amat[14] = Amat[15] = 0.0f;
        #pragma unroll
        for (int i = 0; i < 16; ++i) wsA[b*16 + i] = Amat[i];
    }

    // wave32 reduction (B <= 32; inactive lanes carry zeros)
    #pragma unroll
    for (int off = 16; off > 0; off >>= 1) {
        st   += __shfl_xor(st,   off, 32);
        rdeg += __shfl_xor(rdeg, off, 32);
    }
    if (b == 0) {
        wsS[0] = st   / (float)B;   // loss_transl (mean over batch)
        wsS[1] = rdeg / (float)B;   // loss_rot    (mean over batch)
    }
}

// ---------------------------------------------------------------------------
// Kernel 2: stream points, 16 per WMMA.  grid = (GRID_X, B), block = 256
// (8 waves).  Deterministic: each block writes one partial sum.
// ---------------------------------------------------------------------------
__global__ __launch_bounds__(256) void k_pc_loss(const float* __restrict__ pc,   // [B][4][N]
                                                 const float* __restrict__ wsA,  // [B][16]
                                                 float* __restrict__ partials,   // [B*GRID_X]
                                                 int N) {
    const int b      = blockIdx.y;
    const int lane   = threadIdx.x & 31;
    const int nwaves = blockDim.x >> 5;
    // Wave id as an SGPR so loop control is scalar (no EXEC churn in the loop,
    // and the tail branch below is a scalar branch -> EXEC stays all-ones,
    // which WMMA requires).
    const int wid_s  = (int)(__builtin_amdgcn_readfirstlane(threadIdx.x) >> 5);
    const int laneN  = lane & 15;   // point within tile / matrix row
    const int half   = lane >> 4;   // 0: K={0,1}, 1: K={2,3}

    const float* pcB = pc + (size_t)b * 4u * (size_t)N;

    float acc = 0.0f;

#if USE_WMMA_F32X4
    // A fragment (16x4, rows 4..15 zero), branchless setup:
    //   VGPR0: lanes0-15 (M=laneN,K=0) | lanes16-31 (M=laneN,K=2)
    //   VGPR1: lanes0-15 (M=laneN,K=1) | lanes16-31 (M=laneN,K=3)
    {
    }
    const bool  rowv = (laneN < 4);
    const int   rowc = rowv ? laneN : 0;
    const float* Ab  = wsA + b * 16;
    float a0 = Ab[rowc*4 + half*2 + 0];
    float a1 = Ab[rowc*4 + half*2 + 1];
    a0 = rowv ? a0 : 0.0f;
    a1 = rowv ? a1 : 0.0f;
    v2f Af; Af[0] = a0; Af[1] = a1;

    // B fragment sources: half 0 loads x (K=0) and y (K=1);
    //                     half 1 loads z (K=2), K=3 row is the constant 1.
    const float* p0 = pcB + (size_t)(half * 2) * (size_t)N;  // x or z row
    const float* p1 = pcB + (size_t)N;                       // y row (upper half
                                                             // re-reads same 64B)
    const int ntfull  = N >> 4;                 // full 16-point tiles
    const int tstride = gridDim.x * nwaves;

    #pragma unroll 2
    for (int t = blockIdx.x * nwaves + wid_s; t < ntfull; t += tstride) {
        const int n = (t << 4) + laneN;
        float b0v = p0[n];
        float yv  = p1[n];
        float b1v = half ? 1.0f : yv;           // single v_cndmask
        v2f Bf; Bf[0] = b0v; Bf[1] = b1v;
        v8f c = {};
        // D(16x16) = A(16x4) x B(4x16); rows 0..2 = delta x/y/z of 16 points,
        // rows 4..15 (incl. upper-lane rows 8..10) are exact zeros.
        v8f d = __builtin_amdgcn_wmma_f32_16x16x4_f32(
            false, Af, false, Bf, (short)0, c, false, false);
        float s2 = d[0]*d[0] + d[1]*d[1] + d[2]*d[2];
        acc += fast_sqrtf(s2);                  // upper lanes add sqrt(0) = 0
    }

    // Ragged tail (scalar branch; EXEC remains all-ones for the WMMA).
    if ((N & 15) != 0 && blockIdx.x == 0 && wid_s == 0) {
        const int n  = (ntfull << 4) + laneN;
        const bool inb = (n < N);
        const int  nc  = inb ? n : 0;
        float t0 = p0[nc];
        float t1 = p1[nc];
        float b0v = inb ? t0 : 0.0f;
        float yv  = inb ? t1 : 0.0f;
        float onev = inb ? 1.0f : 0.0f;
        float b1v = half ? onev : yv;
        v2f Bf; Bf[0] = b0v; Bf[1] = b1v;
        v8f c = {};
        v8f d = __builtin_amdgcn_wmma_f32_16x16x4_f32(
            false, Af, false, Bf, (short)0, c, false, false);
        float s2 = d[0]*d[0] + d[1]*d[1] + d[2]*d[2];
        acc += fast_sqrtf(s2);
    }
#else
    // VALU fallback: one point per lane, uniform A (scalarizable loads).
    const float* Ab = wsA + b * 16;
    float A00 = Ab[0],  A01 = Ab[1],  A02 = Ab[2],  A03 = Ab[3];
    float A10 = Ab[4],  A11 = Ab[5],  A12 = Ab[6],  A13 = Ab[7];
    float A20 = Ab[8],  A21 = Ab[9],  A22 = Ab[10], A23 = Ab[11];
    for (int n = blockIdx.x * blockDim.x + threadIdx.x; n < N;
         n += gridDim.x * blockDim.x) {
        float x = pcB[n], y = pcB[N + n], z = pcB[2*(size_t)N + n];
        float dx = fmaf(A00, x, fmaf(A01, y, fmaf(A02, z, A03)));
        float dy = fmaf(A10, x, fmaf(A11, y, fmaf(A12, z, A13)));
        float dz = fmaf(A20, x, fmaf(A21, y, fmaf(A22, z, A23)));
        acc += fast_sqrtf(dx*dx + dy*dy + dz*dz);
    }
#endif

    // wave reduction
    #pragma unroll
    for (int off = 16; off > 0; off >>= 1) acc += __shfl_xor(acc, off, 32);

    __shared__ float ssum[8];
    if (lane == 0) ssum[wid_s] = acc;
    __syncthreads();
    if (threadIdx.x == 0) {
        float s = 0.0f;
        for (int i = 0; i < nwaves; ++i) s += ssum[i];
        partials[(size_t)b * gridDim.x + blockIdx.x] = s;
    }
}

// ---------------------------------------------------------------------------
// Kernel 3: deterministic final reduction + loss combination.
// out = [total, loss_transl, loss_rot, point_clouds_loss/B]
// ---------------------------------------------------------------------------
__global__ __launch_bounds__(256) void k_finalize(const float* __restrict__ partials, int P,
                                                  const float* __restrict__ wsS,
                                                  float* __restrict__ out, int N, int B) {
    __shared__ float s[256];
    float a = 0.0f;
    for (int i = threadIdx.x; i < P; i += blockDim.x) a += partials[i];
    s[threadIdx.x] = a;
    __syncthreads();
    #pragma unroll
    for (int off = 128; off > 0; off >>= 1) {
        if (threadIdx.x < off) s[threadIdx.x] += s[threadIdx.x + off];
        __syncthreads();
    }
    if (threadIdx.x == 0) {
        float pcl   = s[0] / (float)N;       // sum_b mean_n err
        float lt    = wsS[0];
        float lr    = wsS[1];
        float pose  = 1.0f * lt + 1.0f * lr; // RESCALE_TRANS = RESCALE_ROT = 1
        float pclB  = pcl / (float)B;
        out[0] = 0.5f * pose + 0.5f * pclB;  // WEIGHT_PC = 0.5
        out[1] = lt;
        out[2] = lr;
        out[3] = pclB;
    }
}

// ---------------------------------------------------------------------------
extern "C" void kernel_launch(void* const* d_in, const int* in_sizes, int n_in,
                              void* d_out, int out_size, void* d_ws, size_t ws_size,
                              hipStream_t stream) {
    const float* pc  = (const float*)d_in[0];  // [B,4,N]
    const float* ttr = (const float*)d_in[1];  // [B,3] target_transl
    const float* tro = (const float*)d_in[2];  // [B,4] target_rot
    const float* ptr = (const float*)d_in[3];  // [B,3] transl_err
    const float* pro = (const float*)d_in[4];  // [B,4] rot_err

    const int B = in_sizes[1] / 3;
    const int N = in_sizes[0] / (4 * B);

    float* ws       = (float*)d_ws;
    float* wsA      = ws;              // B*16 floats
    float* wsS      = ws + B * 16;     // 2 floats
    float* partials = ws + B * 16 + 2; // B*GRID_X floats

    k_pose_and_mats<<<1, 32, 0, stream>>>(ttr, tro, ptr, pro, wsA, wsS, B);

    dim3 g2(GRID_X, B);
    k_pc_loss<<<g2, 256, 0, stream>>>(pc, wsA, partials, N);

    k_finalize<<<1, 256, 0, stream>>>(partials, GRID_X * B, wsS,
                                      (float*)d_out, N, B);
}